// Atlas_28836410425820
// MI455X (gfx1250) — compile-verified
//
#include <hip/hip_runtime.h>

#define NX 128
#define NY 128
#define NZ 64
#define NVOX (NX*NY*NZ)        // 1048576 voxels per batch
#define NB 2
#define NC 32
#define FH 120
#define FW 160
#define FHW (FH*FW)            // 19200
#define VOXSZ 0.04f

typedef __attribute__((ext_vector_type(2))) float v2f;
typedef __attribute__((ext_vector_type(4))) float v4f;
typedef __attribute__((ext_vector_type(8))) float v8f;

// out layout: volume [NB][NC][NVOX] (67,108,864 f32) then valid [NB][NVOX] (2,097,152 f32)
__global__ __launch_bounds__(256) void atlas_backproject(
    const float* __restrict__ proj,    // [NB,3,4]
    const float* __restrict__ feat,    // [NB,NC,FH,FW]
    const float* __restrict__ origin,  // [NB,3]
    float* __restrict__ out)
{
    const int t    = blockIdx.x * blockDim.x + threadIdx.x;  // 524288 threads total
    const int lane = threadIdx.x & 31;
    const int b    = t >> 18;                 // 2^18 threads per batch (wave-uniform)
    const int v4   = (t & 0x3FFFF) << 2;      // this lane's first voxel (4 consecutive iz)
    const int V0   = ((t & ~31) & 0x3FFFF) << 2;  // wave base voxel (wave-uniform)

    const float* P = proj + b * 12;
    const float ox = origin[b*3+0], oy = origin[b*3+1], oz = origin[b*3+2];

    // ---- A matrix (16x4 f32): proj rows in M=0..2 AND duplicated in M=8..10 ----
    // A layout: lanes 0-15 = M, vgpr0 holds K=0 (lanes<16) / K=2 (lanes>=16), vgpr1 K=1/K=3.
    const int  m    = lane & 15;
    const int  r    = m & 7;                  // rows {0,1,2,8,9,10} -> proj row r
    const bool arow = (r < 3);
    const int  k0   = (lane < 16) ? 0 : 2;
    v2f a;
    a.x = arow ? P[r*4 + k0]     : 0.0f;
    a.y = arow ? P[r*4 + k0 + 1] : 0.0f;

    // ---- B matrices (4x16 f32): column N = world coords of voxel V0 + g*64 + 4*N ----
    // B layout mirrors A transposed: lanes 0-15 hold K=0(v0)/K=1(v1); lanes 16-31 K=2/K=3.
    v2f bm[2];
#pragma unroll
    for (int g = 0; g < 2; ++g) {
        const int vb = V0 + g*64 + 4*m;
        const float wx = (float)(vb >> 13)        * VOXSZ + ox;
        const float wy = (float)((vb >> 6) & 127) * VOXSZ + oy;
        const float wz = (float)(vb & 63)         * VOXSZ + oz;
        bm[g].x = (lane < 16) ? wx : wz;      // K=0 row: wx ; K=2 row: wz
        bm[g].y = (lane < 16) ? wy : 1.0f;    // K=1 row: wy ; K=3 row: 1
    }

    // D = A x B : camera rows 0..2 land in d[0..2] for lanes 0-15 (M=0..2)
    // and, via the duplicated rows, for lanes 16-31 too (M=8..10).
    v8f cz = {};
    v8f d0 = __builtin_amdgcn_wmma_f32_16x16x4_f32(false, a, false, bm[0],
                                                   (short)0, cz, false, false);
    v8f d1 = __builtin_amdgcn_wmma_f32_16x16x4_f32(false, a, false, bm[1],
                                                   (short)0, cz, false, false);

    // WMMA#0 covers voxels V0+4*N (owned by lanes 0-15);
    // WMMA#1 covers V0+64+4*N = V0+4*L for lanes L>=16.
    float cam0 = (lane < 16) ? d0[0] : d1[0];
    float cam1 = (lane < 16) ? d0[1] : d1[1];
    float cam2 = (lane < 16) ? d0[2] : d1[2];

    // camera delta for +1 step in iz
    const float dcx = P[2]  * VOXSZ;
    const float dcy = P[6]  * VOXSZ;
    const float dcz = P[10] * VOXSZ;

    int   idx[4];
    float vf[4];
#pragma unroll
    for (int j = 0; j < 4; ++j) {
        const float c0 = cam0 + (float)j * dcx;
        const float c1 = cam1 + (float)j * dcy;
        const float c2 = cam2 + (float)j * dcz;
        const float fpx = rintf(c0 / c2);     // round-half-even == jnp.round
        const float fpy = rintf(c1 / c2);
        const bool ok = (fpx >= 0.0f) && (fpx <= (float)(FW-1)) &&
                        (fpy >= 0.0f) && (fpy <= (float)(FH-1)) && (c2 > 0.0f);
        const float cx = fminf(fmaxf(fpx, 0.0f), (float)(FW-1));  // NaN-safe clamp
        const float cy = fminf(fmaxf(fpy, 0.0f), (float)(FH-1));
        idx[j] = (int)cy * FW + (int)cx;      // always in-bounds
        vf[j]  = ok ? 1.0f : 0.0f;
    }

    const float* fb = feat + (size_t)b * NC * FHW;
    __builtin_prefetch(fb + idx[0], 0, 0);    // global_prefetch_b8
    __builtin_prefetch(fb + idx[3], 0, 0);

    // gather 32 channels; 16B coalesced stores (512B contiguous per wave).
    // Stores are write-once streaming data -> non-temporal hint, so the 264MB
    // output stream doesn't evict the L2-resident 4.9MB feature tensor.
#pragma unroll 4
    for (int c = 0; c < NC; ++c) {
        const float* fc = fb + c * FHW;
        const float g0 = fc[idx[0]];
        const float g1 = fc[idx[1]];
        const float g2 = fc[idx[2]];
        const float g3 = fc[idx[3]];
        v4f g;
        g.x = (vf[0] != 0.0f) ? g0 : 0.0f;
        g.y = (vf[1] != 0.0f) ? g1 : 0.0f;
        g.z = (vf[2] != 0.0f) ? g2 : 0.0f;
        g.w = (vf[3] != 0.0f) ? g3 : 0.0f;
        __builtin_nontemporal_store(g, (v4f*)(out + ((size_t)(b * NC + c)) * NVOX + v4));
    }

    // valid mask output (as float 1.0/0.0), after the volume block
    v4f vv = { vf[0], vf[1], vf[2], vf[3] };
    __builtin_nontemporal_store(vv, (v4f*)(out + (size_t)NB * NC * NVOX + (size_t)b * NVOX + v4));
}

extern "C" void kernel_launch(void* const* d_in, const int* in_sizes, int n_in,
                              void* d_out, int out_size, void* d_ws, size_t ws_size,
                              hipStream_t stream) {
    const float* proj   = (const float*)d_in[0];   // projection [2,3,4]
    const float* feat   = (const float*)d_in[1];   // features   [2,32,120,160]
    const float* origin = (const float*)d_in[2];   // origin     [2,3]
    float* out = (float*)d_out;

    const int total_threads = (NB * NVOX) / 4;     // 524288
    dim3 block(256);
    dim3 grid(total_threads / 256);                // 2048, exact cover (EXEC all-1s for WMMA)
    atlas_backproject<<<grid, block, 0, stream>>>(proj, feat, origin, out);
}